// surrogate_gradient_adder_dense_28905129902467
// MI455X (gfx1250) — compile-verified
//
#include <hip/hip_runtime.h>
#include <stdint.h>

#define THREADS 256

typedef float f32x4 __attribute__((ext_vector_type(4)));

// Non-temporal 128-bit store: output is write-once, keep it out of L2 so the
// (re-read every replay) 64 MiB input stays L2-resident (L2 = 192 MB).
__device__ __forceinline__ void store_nt4(float* p, f32x4 v) {
    __builtin_nontemporal_store(v, (f32x4*)p);
}

__device__ __forceinline__ f32x4 bits4(unsigned v, int k) {
    f32x4 r;
    r.x = (float)((v >> k) & 1u);
    r.y = (float)((v >> (k + 1)) & 1u);
    r.z = (float)((v >> (k + 2)) & 1u);
    r.w = (float)((v >> (k + 3)) & 1u);
    return r;
}

__global__ __launch_bounds__(THREADS)
void surrogate_adder_kernel(const float* __restrict__ x,
                            float* __restrict__ out_sum,
                            float* __restrict__ out_carry,
                            int nrows /* batch*8 sub-rows */) {
    __shared__ float smem[THREADS * 16];   // 16 KB: 64 B staged per thread

    const int g = blockIdx.x * THREADS + threadIdx.x;   // sub-row = i*8 + j
    const bool active = g < nrows;                      // nrows % 8 == 0, so all
                                                        // lanes of an 8-group agree

    // ---- CDNA5 async global->LDS staging (ASYNCcnt path) -------------------
    // Each lane copies its 64 B (16 floats) with 4 x b128 async transfers.
    // INST_OFFSET applies to both the LDS and global addresses, so one base
    // LDS-offset VGPR + one 64-bit global-address VGPR pair suffice.
    const unsigned ldsOff = (unsigned)(uintptr_t)(&smem[threadIdx.x * 16]);
    if (active) {
        const unsigned long long ga =
            (unsigned long long)(uintptr_t)(x + (size_t)g * 16);
        asm volatile(
            "global_load_async_to_lds_b128 %0, %1, off\n\t"
            "global_load_async_to_lds_b128 %0, %1, off offset:16\n\t"
            "global_load_async_to_lds_b128 %0, %1, off offset:32\n\t"
            "global_load_async_to_lds_b128 %0, %1, off offset:48"
            :: "v"(ldsOff), "v"(ga) : "memory");
    }
    // Wait for our own async transfers; each thread consumes only its own
    // staged bytes, so no workgroup barrier is required.
#if __has_builtin(__builtin_amdgcn_s_wait_asynccnt)
    __builtin_amdgcn_s_wait_asynccnt(0);
    asm volatile("" ::: "memory");
#else
    asm volatile("s_wait_asynccnt 0x0" ::: "memory");
#endif

    if (!active) return;

    const float4* sv = (const float4*)(&smem[threadIdx.x * 16]);
    float4 v0 = sv[0];
    float4 v1 = sv[1];
    float4 v2 = sv[2];
    float4 v3 = sv[3];

    // ---- pack 16 bits -> integer, exact in f32 (value <= 65535 < 2^24) ----
    float p0 = v0.x;                       // bits 0..7
    p0 = fmaf(v0.y,     2.0f, p0);
    p0 = fmaf(v0.z,     4.0f, p0);
    p0 = fmaf(v0.w,     8.0f, p0);
    p0 = fmaf(v1.x,    16.0f, p0);
    p0 = fmaf(v1.y,    32.0f, p0);
    p0 = fmaf(v1.z,    64.0f, p0);
    p0 = fmaf(v1.w,   128.0f, p0);
    float p1 = v2.x * 256.0f;              // bits 8..15 (parallel chain)
    p1 = fmaf(v2.y,   512.0f, p1);
    p1 = fmaf(v2.z,  1024.0f, p1);
    p1 = fmaf(v2.w,  2048.0f, p1);
    p1 = fmaf(v3.x,  4096.0f, p1);
    p1 = fmaf(v3.y,  8192.0f, p1);
    p1 = fmaf(v3.z, 16384.0f, p1);
    p1 = fmaf(v3.w, 32768.0f, p1);
    const unsigned a = (unsigned)(p0 + p1);

    // ---- x_sum over the 8 sub-rows of this batch row (lanes j=0..7) -------
    int s = (int)a;
    s += __shfl_xor(s, 1);
    s += __shfl_xor(s, 2);
    s += __shfl_xor(s, 4);

    // ---- exact ripple-carry results via integer identities -----------------
    // b = low 16 bits of residual; t = a + b; carry-out_k = bit k of (a^b^t)>>1
    const unsigned b  = ((unsigned)s - a) & 0xFFFFu;
    const unsigned t  = a + b;
    const unsigned cv = (a ^ b ^ t) >> 1;

    // ---- expand to floats, 128-bit non-temporal vector stores --------------
    float* o = out_sum + (size_t)g * 16;
    store_nt4(o +  0, bits4(t, 0));
    store_nt4(o +  4, bits4(t, 4));
    store_nt4(o +  8, bits4(t, 8));
    store_nt4(o + 12, bits4(t, 12));

    float* c = out_carry + (size_t)g * 16;
    store_nt4(c +  0, bits4(cv, 0));
    store_nt4(c +  4, bits4(cv, 4));
    store_nt4(c +  8, bits4(cv, 8));
    store_nt4(c + 12, bits4(cv, 12));
}

extern "C" void kernel_launch(void* const* d_in, const int* in_sizes, int n_in,
                              void* d_out, int out_size, void* d_ws, size_t ws_size,
                              hipStream_t stream) {
    const float* x = (const float*)d_in[0];
    float* out = (float*)d_out;

    const int total_in = in_sizes[0];        // batch * 8 * 16
    const int nrows    = total_in / 16;      // batch * 8 sub-rows
    float* out_sum   = out;
    float* out_carry = out + (size_t)nrows * 16;

    const int blocks = (nrows + THREADS - 1) / THREADS;
    hipLaunchKernelGGL(surrogate_adder_kernel, dim3(blocks), dim3(THREADS), 0,
                       stream, x, out_sum, out_carry, nrows);
}